// GraphEncoder_8761733284233
// MI455X (gfx1250) — compile-verified
//
#include <hip/hip_runtime.h>
#include <math.h>

typedef __attribute__((ext_vector_type(2))) float v2f;
typedef __attribute__((ext_vector_type(8))) float v8f;

#define N_NODES 50000
#define N_EDGES 500000
#define IN_CH   128
#define OUT_CH  256

// ---------------------------------------------------------------------------
// 1) deg[n] = 1.0 (self-loop weight); re-run every call for replay safety.
// ---------------------------------------------------------------------------
__global__ void __launch_bounds__(256) init_deg_kernel(float* __restrict__ deg) {
    int n = blockIdx.x * blockDim.x + threadIdx.x;
    if (n < N_NODES) deg[n] = 1.0f;
}

// ---------------------------------------------------------------------------
// 2) sw[e] = sigmoid(ew[e]); deg[col[e]] += sw[e]  (atomic)
// ---------------------------------------------------------------------------
__global__ void __launch_bounds__(256) edge_prep_kernel(const int* __restrict__ ei,
                                                        const float* __restrict__ ew,
                                                        float* __restrict__ sw,
                                                        float* __restrict__ deg) {
    int e = blockIdx.x * blockDim.x + threadIdx.x;
    if (e >= N_EDGES) return;
    float w = 1.0f / (1.0f + __expf(-ew[e]));
    sw[e] = w;
    int col = ei[N_EDGES + e];
    atomicAdd(&deg[col], w);
}

// ---------------------------------------------------------------------------
// 3) dinv[n] = deg>0 ? rsqrt(deg) : 0   (in place)
// ---------------------------------------------------------------------------
__global__ void __launch_bounds__(256) dinv_kernel(float* __restrict__ deg) {
    int n = blockIdx.x * blockDim.x + threadIdx.x;
    if (n >= N_NODES) return;
    float d = deg[n];
    deg[n] = (d > 0.0f) ? rsqrtf(d) : 0.0f;
}

// ---------------------------------------------------------------------------
// 4) h = x @ W via V_WMMA_F32_16X16X4_F32.
//    One wave per 16x16 output tile; K-loop over 128 in steps of 4 (32 WMMAs).
//    A 16x4 layout: VGPR0 = {K=k0 (lanes 0-15), K=k0+2 (lanes 16-31)},
//                   VGPR1 = {K=k0+1, K=k0+3}  -> contiguous 8B load per lane.
//    B 4x16: VGPR0 = row k0 / k0+2 striped over half-waves, VGPR1 = +1 row.
//    C/D 16x16 f32: VGPR r -> row m0 + r + 8*half, col n0 + (lane&15).
// ---------------------------------------------------------------------------
__global__ void __launch_bounds__(128) gemm_wmma_kernel(const float* __restrict__ x,
                                                        const float* __restrict__ W,
                                                        float* __restrict__ h) {
    int wave  = blockIdx.x * (blockDim.x >> 5) + (threadIdx.x >> 5);
    int tileM = wave >> 4;          // 3125 row tiles
    int tileN = wave & 15;          // 16 col tiles
    if (tileM >= N_NODES / 16) return;

    int lane = threadIdx.x & 31;
    int half = lane >> 4;           // 0: K pair (0,1); 1: K pair (2,3)
    int lm   = lane & 15;
    int m0   = tileM * 16;
    int n0   = tileN * 16;

    const float* aBase = x + (size_t)(m0 + lm) * IN_CH + half * 2;
    const float* bBase = W + (size_t)(half * 2) * OUT_CH + n0 + lm;

    v8f c = {};
#pragma unroll
    for (int k0 = 0; k0 < IN_CH; k0 += 4) {
        float2 av = *(const float2*)(aBase + k0);       // global_load_b64
        v2f a; a.x = av.x; a.y = av.y;
        v2f b;
        b.x = bBase[(size_t)k0 * OUT_CH];               // row k0 + 2*half
        b.y = bBase[(size_t)(k0 + 1) * OUT_CH];         // row k0 + 2*half + 1
        // 8 args: (neg_a, A, neg_b, B, c_mod, C, reuse_a, reuse_b)
        c = __builtin_amdgcn_wmma_f32_16x16x4_f32(false, a, false, b,
                                                  (short)0, c, false, false);
    }

    float* op = h + (size_t)(m0 + half * 8) * OUT_CH + n0 + lm;
#pragma unroll
    for (int r = 0; r < 8; ++r) op[(size_t)r * OUT_CH] = c[r];
}

// ---------------------------------------------------------------------------
// 5) out[n][c] = b[c] + dinv[n]^2 * h[n][c]   (bias + self-loop contribution)
//    float4-vectorized: one thread per 4 channels.
// ---------------------------------------------------------------------------
__global__ void __launch_bounds__(256) out_init_kernel(const float* __restrict__ h,
                                                       const float* __restrict__ dinv,
                                                       const float* __restrict__ bias,
                                                       float* __restrict__ out) {
    int idx = blockIdx.x * blockDim.x + threadIdx.x;    // over N_NODES * 64
    if (idx >= N_NODES * (OUT_CH / 4)) return;
    int n  = idx >> 6;
    int c4 = idx & 63;
    float s = dinv[n];
    s = s * s;
    float4 hv = ((const float4*)h)[idx];
    float4 bv = ((const float4*)bias)[c4];
    float4 ov;
    ov.x = bv.x + s * hv.x;
    ov.y = bv.y + s * hv.y;
    ov.z = bv.z + s * hv.z;
    ov.w = bv.w + s * hv.w;
    ((float4*)out)[idx] = ov;
}

// ---------------------------------------------------------------------------
// 6) Edge scatter: out[col] += (dinv[row]*sw[e]*dinv[col]) * h[row]
//    4 edges per 256-thread block; each thread handles 4 channels (float4
//    gather of h, 4x global_atomic_add_f32 into out).
// ---------------------------------------------------------------------------
__global__ void __launch_bounds__(256) scatter_kernel(const int* __restrict__ ei,
                                                      const float* __restrict__ sw,
                                                      const float* __restrict__ dinv,
                                                      const float* __restrict__ h,
                                                      float* __restrict__ out) {
    int e = blockIdx.x * 4 + (threadIdx.x >> 6);
    if (e >= N_EDGES) return;
    int t = threadIdx.x & 63;       // channel group: 4 channels each

    int r = ei[e];
    int c = ei[N_EDGES + e];
    float norm = dinv[r] * sw[e] * dinv[c];

    float4 hv = ((const float4*)(h + (size_t)r * OUT_CH))[t];  // global_load_b128
    float* op = out + (size_t)c * OUT_CH + t * 4;
    atomicAdd(op + 0, norm * hv.x);
    atomicAdd(op + 1, norm * hv.y);
    atomicAdd(op + 2, norm * hv.z);
    atomicAdd(op + 3, norm * hv.w);
}

// ---------------------------------------------------------------------------
extern "C" void kernel_launch(void* const* d_in, const int* in_sizes, int n_in,
                              void* d_out, int out_size, void* d_ws, size_t ws_size,
                              hipStream_t stream) {
    const float* x    = (const float*)d_in[0];   // [50000,128]
    const float* W    = (const float*)d_in[1];   // [128,256]
    const float* bias = (const float*)d_in[2];   // [256]
    const int*   ei   = (const int*)  d_in[3];   // [2,500000]
    const float* ew   = (const float*)d_in[4];   // [500000]
    float*       out  = (float*)d_out;           // [50000,256]

    // scratch layout: h (12.8M floats) | deg/dinv (50000) | sw (500000)
    float* h   = (float*)d_ws;
    float* deg = h + (size_t)N_NODES * OUT_CH;
    float* sw  = deg + N_NODES;

    init_deg_kernel<<<(N_NODES + 255) / 256, 256, 0, stream>>>(deg);
    edge_prep_kernel<<<(N_EDGES + 255) / 256, 256, 0, stream>>>(ei, ew, sw, deg);
    dinv_kernel<<<(N_NODES + 255) / 256, 256, 0, stream>>>(deg);

    // 50000 tiles, 4 waves per 128-thread block
    gemm_wmma_kernel<<<(N_NODES / 16) * 16 / 4, 128, 0, stream>>>(x, W, h);

    out_init_kernel<<<(N_NODES * (OUT_CH / 4) + 255) / 256, 256, 0, stream>>>(h, deg, bias, out);
    scatter_kernel<<<N_EDGES / 4, 256, 0, stream>>>(ei, sw, deg, h, out);
}